// SimDense_85847806313217
// MI455X (gfx1250) — compile-verified
//
#include <hip/hip_runtime.h>
#include <stdint.h>

#define RHO_F   1060.0f
#define DX_F    0.001f
#define CCFL_F  0.9f
#define EPS_AF  1e-6f

#define BLOCK   256
#define CPT     4                 // cells per thread
#define CPB     (BLOCK*CPT)       // 1024 cells per block

// workspace layout (32-bit words)
#define WS_CNT   0                // grid-barrier counter
#define WS_SMAX  256              // per-step smax slots (float bits, uint atomicMax)
#define WS_HALO  1536             // ping-pong halo buffers start here
#define MAX_T    1024

__global__ void init_ws_kernel(unsigned* ws, int nwords) {
  int i = blockIdx.x * blockDim.x + threadIdx.x;
  if (i < nwords) ws[i] = 0u;
}

__device__ __forceinline__ float softplusf(float x) {
  // jax.nn.softplus, numerically stable form
  return fmaxf(x, 0.0f) + log1pf(expf(-fabsf(x)));
}

__device__ __forceinline__ void grid_sync(unsigned* cnt, unsigned& target, unsigned nb) {
  __threadfence();                 // release this thread's global writes
  __syncthreads();
  target += nb;
  if (threadIdx.x == 0) {
    atomicAdd(cnt, 1u);
    while (atomicAdd(cnt, 0u) < target) {
      __builtin_amdgcn_s_sleep(1);
    }
  }
  __syncthreads();
  __threadfence();                 // acquire: invalidate WGP$ before consuming
}

__global__ __launch_bounds__(BLOCK) void fv_solver(
    const float* __restrict__ Rs,
    const float* __restrict__ sim_dat,        // (2, M)
    const float* __restrict__ sim_dat_const,  // (11, M)
    const float* __restrict__ input_data,     // (T)
    const int*   __restrict__ strides,        // (3, 2)
    const int*   __restrict__ upper_p,        // scalar
    float*       __restrict__ out,            // (T, 3)
    unsigned*    __restrict__ ws,
    int M)
{
  __shared__ float sA[CPB + 2];
  __shared__ float sQ[CPB + 2];
  __shared__ float sF[CPB + 2];
  __shared__ unsigned sMax;

  const int b        = blockIdx.x;
  const int nb       = gridDim.x;
  const int tid      = threadIdx.x;
  const int gbase    = b * CPB + tid * CPT;
  const int lastLoc  = min(CPB, M - b * CPB) - 1;   // last valid local cell

  unsigned* cnt      = ws + WS_CNT;
  unsigned* smaxSlot = ws + WS_SMAX;
  float*    haloBase = (float*)(ws + WS_HALO);
  const int haloStride = (nb + 2) * 6;              // per ping-pong buffer

  int T = *upper_p;
  if (T > MAX_T) T = MAX_T;

  const int s00 = strides[0], s01 = strides[1];
  const int s10 = strides[2], s11 = strides[3];
  const int s20 = strides[4], s21 = strides[5];

  // ---- per-cell state & constants live in registers for the whole run ----
  float A[CPT], Q[CPT], A0v[CPT], betav[CPT], cbv[CPT], Kv[CPT], Rt[CPT], Pj[CPT];
  bool  valid[CPT], isStart[CPT], isEnd[CPT];
  int   midv[CPT];

  #pragma unroll
  for (int j = 0; j < CPT; ++j) {
    const int i = gbase + j;
    valid[j] = (i < M);
    isStart[j] = false; isEnd[j] = false; midv[j] = -1;
    Rt[j] = 1.0f; A[j] = 1.0f; Q[j] = 0.0f;
    A0v[j] = 1.0f; betav[j] = 1.0f; cbv[j] = 1.0f; Kv[j] = 1.0f; Pj[j] = 0.0f;
    if (valid[j]) {
      A[j] = sim_dat[i];
      Q[j] = 0.1f * sim_dat[M + i];
      const float a0 = sim_dat_const[i] + 0.5f;        // A0
      const float be = sim_dat_const[M + i] + 1.0f;    // beta
      A0v[j]  = a0;
      betav[j]= be;
      cbv[j]  = (0.5f * be) / RHO_F;                   // for c = sqrt(cb*sq)
      Kv[j]   = be / ((3.0f * RHO_F) * a0);            // for F_Q
      isStart[j] = (i == s00) || (i == s10) || (i == s20);
      isEnd[j]   = (i == s01 - 1) || (i == s11 - 1) || (i == s21 - 1);
      if      (i == (s00 + s01) / 2) midv[j] = 0;
      else if (i == (s10 + s11) / 2) midv[j] = 1;
      else if (i == (s20 + s21) / 2) midv[j] = 2;
      if (isEnd[j]) {
        float r1 = sim_dat_const[7 * M + i] + 0.5f;
        float r2 = sim_dat_const[8 * M + i] + 0.5f;
        const float sp0 = softplusf(Rs[0]);
        const float sp1 = softplusf(Rs[1]);
        const float sp2 = softplusf(Rs[2]);
        const float sp3 = softplusf(Rs[3]);
        const bool m1 = (i >= s10) && (i < s11);
        const bool m2 = (i >= s20) && (i < s21);
        if (m1)      { r1 = sp0 * r1; r2 = sp1 * r2; }
        else if (m2) { r1 = sp2 * r1; r2 = sp3 * r2; }
        Rt[j] = r1 + r2;
      }
    }
  }

  unsigned barTarget = 0;

  for (int t = 0; t < T; ++t) {
    float* halo = haloBase + (t & 1) * haloStride;    // ping-pong: 1 barrier/step

    if (tid == 0) sMax = 0u;
    __syncthreads();   // also protects LDS reuse vs previous step's phase 2

    // ---------------- phase 1: pointwise + local reduction ----------------
    float lm = 0.0f;
    #pragma unroll
    for (int j = 0; j < CPT; ++j) {
      if (valid[j]) {
        const float a  = fmaxf(A[j], EPS_AF);
        A[j] = a;
        const float sq = sqrtf(a / A0v[j]);
        const float P  = betav[j] * (sq - 1.0f);
        const float u  = Q[j] / a;
        const float cc = sqrtf(cbv[j] * sq);
        lm = fmaxf(lm, fabsf(u) + cc);
        const float F  = Q[j] * u + (Kv[j] * a) * sq;  // F_Q ; F_A == Q
        const int loc = tid * CPT + j;
        sA[loc + 1] = a;
        sQ[loc + 1] = Q[j];
        sF[loc + 1] = F;
        Pj[j] = P;
        if (midv[j] >= 0) out[t * 3 + midv[j]] = P;    // scan output
      }
    }
    // wave32 butterfly max, then one LDS atomic per wave, one global per block
    #pragma unroll
    for (int off = 16; off > 0; off >>= 1)
      lm = fmaxf(lm, __shfl_xor(lm, off, 32));
    if ((tid & 31) == 0) atomicMax(&sMax, __float_as_uint(lm));
    __syncthreads();

    if (tid == 0) {
      atomicMax(&smaxSlot[t], sMax);                   // positive floats: uint-max ok
      halo[(b + 1) * 6 + 0] = sA[1];                   // block's first cell
      halo[(b + 1) * 6 + 1] = sQ[1];
      halo[(b + 1) * 6 + 2] = sF[1];
    } else if (tid == 1) {
      halo[(b + 1) * 6 + 3] = sA[lastLoc + 1];         // block's last cell
      halo[(b + 1) * 6 + 4] = sQ[lastLoc + 1];
      halo[(b + 1) * 6 + 5] = sF[lastLoc + 1];
    }

    grid_sync(cnt, barTarget, nb);   // the single device-wide sync per step

    // -------- phase 2: halo fetch (CDNA5 async global->LDS), flux update --
    if (tid < 6) {
      const float* gsrc;
      float* ldst;
      if      (tid == 0) { gsrc = &halo[b * 6 + 3];       ldst = &sA[0]; }
      else if (tid == 1) { gsrc = &halo[b * 6 + 4];       ldst = &sQ[0]; }
      else if (tid == 2) { gsrc = &halo[b * 6 + 5];       ldst = &sF[0]; }
      else if (tid == 3) { gsrc = &halo[(b + 2) * 6 + 0]; ldst = &sA[CPB + 1]; }
      else if (tid == 4) { gsrc = &halo[(b + 2) * 6 + 1]; ldst = &sQ[CPB + 1]; }
      else               { gsrc = &halo[(b + 2) * 6 + 2]; ldst = &sF[CPB + 1]; }
      const unsigned lds_off = (unsigned)(uintptr_t)ldst;               // flat LDS addr[31:0]
      const unsigned long long gaddr = (unsigned long long)(uintptr_t)gsrc;
      asm volatile("global_load_async_to_lds_b32 %0, %1, off"
                   :: "v"(lds_off), "v"(gaddr) : "memory");
    }
    const float smax = __uint_as_float(smaxSlot[t]);
    asm volatile("s_wait_asynccnt 0" ::: "memory");
    __syncthreads();

    const float dt  = CCFL_F * DX_F / smax;
    const float lam = dt / DX_F;
    const float h   = 0.5f * smax;

    #pragma unroll
    for (int j = 0; j < CPT; ++j) {
      const int i = gbase + j;
      if (valid[j]) {
        float An = A[j], Qn = Q[j];
        if (i > 0 && i < M - 1) {
          const int lc = tid * CPT + j;
          const float aL = sA[lc], aC = sA[lc + 1], aR = sA[lc + 2];
          const float qL = sQ[lc], qC = sQ[lc + 1], qR = sQ[lc + 2];
          const float fL = sF[lc], fC = sF[lc + 1], fR = sF[lc + 2];
          const float fhA_l = 0.5f * (qL + qC) - h * (aC - aL);
          const float fhA_r = 0.5f * (qC + qR) - h * (aR - aC);
          const float fhQ_l = 0.5f * (fL + fC) - h * (qC - qL);
          const float fhQ_r = 0.5f * (fC + fR) - h * (qR - qC);
          An = aC - lam * (fhA_r - fhA_l);
          Qn = qC - lam * (fhQ_r - fhQ_l);
        }
        if (isStart[j]) Qn = input_data[t];      // inflow BC
        if (isEnd[j])   Qn = Pj[j] / Rt[j];      // outflow BC
        A[j] = fmaxf(An, EPS_AF);
        Q[j] = Qn;
      }
    }
    // no second barrier: next step writes the other halo buffer & smax slot
  }
}

extern "C" void kernel_launch(void* const* d_in, const int* in_sizes, int n_in,
                              void* d_out, int out_size, void* d_ws, size_t ws_size,
                              hipStream_t stream) {
  const float* Rs         = (const float*)d_in[0];
  const float* sim_dat    = (const float*)d_in[1];
  const float* sdc        = (const float*)d_in[2];
  const float* input_data = (const float*)d_in[3];
  const int*   strides    = (const int*)d_in[4];
  const int*   upper_p    = (const int*)d_in[5];

  const int M  = in_sizes[1] / 2;                         // sim_dat is (2, M)
  const int nb = (M + CPB - 1) / CPB;                     // 88 blocks for M=90000

  const int nwords = WS_HALO + 2 * (nb + 2) * 6;          // counter+slots+ping-pong halos
  init_ws_kernel<<<(nwords + 255) / 256, 256, 0, stream>>>((unsigned*)d_ws, nwords);

  fv_solver<<<nb, BLOCK, 0, stream>>>(Rs, sim_dat, sdc, input_data, strides, upper_p,
                                      (float*)d_out, (unsigned*)d_ws, M);
}